// SingleLrGatLayer_88270167867642
// MI455X (gfx1250) — compile-verified
//
#include <hip/hip_runtime.h>
#include <math.h>

// CDNA5 (gfx1250) fused SingleLrGat layer.
//   out[t,l] = sigmoid( (tr[t,l]*kr[l]+br[l]) * (kl[l]*s[t,l] + bl[l]*suma[t]) )
//   s[t,l]   = sum_m a[t,m] * NL[t,m,l],   suma[t] = sum_m a[t,m]
// t = flattened (b,n), 8192 tiles, each tile NL is 64x64 fp32 = 16 KiB.
// HBM-bound (~134 MiB @ 23.3 TB/s ~= 6 us). Data path: async global->LDS
// copies (ASYNCcnt), reduction on the matrix pipe via V_WMMA_F32_16X16X4_F32
// with the a-vector broadcast into all 16 A rows.

typedef __attribute__((ext_vector_type(2))) float v2f;
typedef __attribute__((ext_vector_type(8))) float v8f;

#define WPB    4        // waves per block
#define TILE_F 4096     // floats per (b,n) tile: 64 (m) x 64 (l)

__global__ __launch_bounds__(WPB * 32)
void lrgat_fused_kernel(const float* __restrict__ tr,   // (8192, 64)
                        const float* __restrict__ nl,   // (8192, 64, 64)
                        const float* __restrict__ a,    // (8192, 64)
                        const float* __restrict__ kl,   // (64)
                        const float* __restrict__ bl,   // (64)
                        const float* __restrict__ kr,   // (64)
                        const float* __restrict__ br,   // (64)
                        float* __restrict__ out)        // (8192, 64)
{
    __shared__ float sm[WPB * TILE_F];   // 64 KiB: one 16 KiB tile per wave

    const int tid  = (int)threadIdx.x;
    const int wave = tid >> 5;
    const int lane = tid & 31;
    const int t    = (int)blockIdx.x * WPB + wave;     // tile id in [0, 8192)

    // ------------------------------------------------------------------
    // Stage the 16 KiB neighbors_ligand tile into LDS with async copies.
    // Each instruction moves 512 B per wave (b128 per lane), 32 issues.
    // ------------------------------------------------------------------
    const float* gsrc  = nl + (size_t)t * TILE_F;                 // wave-uniform -> SGPR pair
    unsigned     laddr = (unsigned)(size_t)(&sm[wave * TILE_F]) + (unsigned)lane * 16u;
    unsigned     voff  = (unsigned)lane * 16u;
#pragma unroll
    for (int i = 0; i < 32; ++i) {
        asm volatile("global_load_async_to_lds_b128 %0, %1, %2"
                     :: "v"(laddr), "v"(voff), "s"(gsrc) : "memory");
        laddr += 512u;
        voff  += 512u;
    }

    const bool hi  = (lane & 16) != 0;
    const int  col = lane & 15;
    const float4* a4 = (const float4*)(a + (size_t)t * 64);

    // Wait for the async tile before touching LDS.
    asm volatile("s_wait_asynccnt 0x0" ::: "memory");

    // ------------------------------------------------------------------
    // s[l] = sum_m a[m]*NL[m,l] on the matrix pipe.
    // A (16x4 f32): row-replicated a[4k..4k+3]  -> lanes<16 hold {K0,K1},
    //               lanes>=16 hold {K2,K3} (ISA 16x4 A layout).
    // B (4x16 f32): lane j=L%16 is column j; V0 = rows {0|2}, V1 = rows {1|3}.
    // D: all 16 rows identical = partial s for that 16-wide l-group.
    // ------------------------------------------------------------------
    v8f c0 = {}, c1 = {}, c2 = {}, c3 = {};
    float sa = 0.0f;
    const float* smw = &sm[wave * TILE_F];
#pragma unroll
    for (int k = 0; k < 16; ++k) {
        float4 q = a4[k];                         // wave-uniform broadcast load
        sa += q.x + q.y + q.z + q.w;
        v2f A;
        A.x = hi ? q.z : q.x;
        A.y = hi ? q.w : q.y;
        const float* rowp = smw + (4 * k + (hi ? 2 : 0)) * 64 + col;
        v2f B0; B0.x = rowp[0];  B0.y = rowp[64];
        v2f B1; B1.x = rowp[16]; B1.y = rowp[80];
        v2f B2; B2.x = rowp[32]; B2.y = rowp[96];
        v2f B3; B3.x = rowp[48]; B3.y = rowp[112];
        c0 = __builtin_amdgcn_wmma_f32_16x16x4_f32(false, A, false, B0, (short)0, c0, false, false);
        c1 = __builtin_amdgcn_wmma_f32_16x16x4_f32(false, A, false, B1, (short)0, c1, false, false);
        c2 = __builtin_amdgcn_wmma_f32_16x16x4_f32(false, A, false, B2, (short)0, c2, false, false);
        c3 = __builtin_amdgcn_wmma_f32_16x16x4_f32(false, A, false, B3, (short)0, c3, false, false);
    }

    // ------------------------------------------------------------------
    // Epilogue: low half-wave owns l-groups 0,1; high half owns 2,3.
    // c[g][0] for lane L is s[g*16 + (L&15)] (row 0 / row 8, rows identical).
    // ------------------------------------------------------------------
    const float s0 = hi ? c2[0] : c0[0];
    const float s1 = hi ? c3[0] : c1[0];
    const int   l0 = (hi ? 32 : 0) + col;
    const int   l1 = l0 + 16;
    const size_t ob = (size_t)t * 64;

    {
        float r = tr[ob + l0] * kr[l0] + br[l0];
        float w = r * (kl[l0] * s0 + bl[l0] * sa);
        out[ob + l0] = 1.0f / (1.0f + __expf(-w));
    }
    {
        float r = tr[ob + l1] * kr[l1] + br[l1];
        float w = r * (kl[l1] * s1 + bl[l1] * sa);
        out[ob + l1] = 1.0f / (1.0f + __expf(-w));
    }
}

extern "C" void kernel_launch(void* const* d_in, const int* in_sizes, int n_in,
                              void* d_out, int out_size, void* d_ws, size_t ws_size,
                              hipStream_t stream) {
    (void)in_sizes; (void)n_in; (void)out_size; (void)d_ws; (void)ws_size;
    const float* tr = (const float*)d_in[0];   // targets_receptor (4,2048,64)
    const float* nl = (const float*)d_in[1];   // neighbors_ligand (4,2048,64,64)
    const float* a  = (const float*)d_in[2];   // a (4,2048,64)
    const float* kl = (const float*)d_in[3];   // kernel_l (64)
    const float* bl = (const float*)d_in[4];   // bias_l (64)
    const float* kr = (const float*)d_in[5];   // kernel_r (64)
    const float* br = (const float*)d_in[6];   // bias_r (64)
    float* out = (float*)d_out;

    const int tiles  = 4 * 2048;               // 8192 (b,n) tiles
    dim3 grid(tiles / WPB);                    // 2048 blocks
    dim3 block(WPB * 32);                      // 4 waves of 32
    lrgat_fused_kernel<<<grid, block, 0, stream>>>(tr, nl, a, kl, bl, kr, br, out);
}